// SheafLayer_83270825935598
// MI455X (gfx1250) — compile-verified
//
#include <hip/hip_runtime.h>
#include <cstddef>

// Problem constants (match reference)
#define NP 64      // patches
#define ND 256     // feature dim (K and output rows)
#define NB 32      // columns
#define LAMBDA_W 0.1f

typedef __attribute__((ext_vector_type(2))) float v2f;
typedef __attribute__((ext_vector_type(8))) float v8f;

// ---------------------------------------------------------------------------
// Zero the workspace accumulator (agg [NP*ND*NB] + loss scalar).
// ---------------------------------------------------------------------------
__global__ __launch_bounds__(256) void ws_zero_kernel(float* __restrict__ w, int n) {
  int idx = blockIdx.x * blockDim.x + threadIdx.x;
  if (idx < n) w[idx] = 0.0f;
}

// ---------------------------------------------------------------------------
// One workgroup per unordered pair (i<j). 256 threads = 8 wave32.
// Wave w owns output rows [32w, 32w+32) x all 32 cols: 2x2 tiles of 16x16.
// Computes M[i,j] = R[i,j] @ F[i] and M[j,i] = R[j,i] @ F[j] with
// v_wmma_f32_16x16x4_f32, accumulates agg atomically, loss via antisymmetric
// diff held entirely in registers.
// ---------------------------------------------------------------------------
__global__ __launch_bounds__(256) void sheaf_pair_kernel(
    const float* __restrict__ F,    // [NP, ND, NB]
    const float* __restrict__ R,    // [NP, NP, ND, ND]
    float* __restrict__ agg,        // [NP, ND, NB] (ws)
    float* __restrict__ loss) {     // scalar (ws)
  const int bx = blockIdx.x;
  const int i = bx >> 6;            // NP = 64
  const int j = bx & 63;
  if (i >= j) return;               // diagonal contributes nothing; i>j mirrored

  const int tid   = threadIdx.x;
  const int lane  = tid & 31;
  const int wave  = tid >> 5;       // 0..7
  const int l16   = lane & 15;
  const int lhalf = lane >> 4;      // 0 or 1
  const int d0    = wave * 32;      // this wave's 32 output rows

  const v8f vzero = {0.f,0.f,0.f,0.f,0.f,0.f,0.f,0.f};
  v8f acc[2][2][2];                 // [dir(ij/ji)][rowtile][coltile]

  #pragma unroll
  for (int t = 0; t < 2; ++t) {
    const int src = t ? j : i;
    const int dst = t ? i : j;
    const float* __restrict__ Rp = R + (size_t)(src * NP + dst) * (ND * ND);
    const float* __restrict__ Fp = F + (size_t)src * (ND * NB);

    // A fragment (16x4 f32): lane row = d0 + 16r + l16, cols = 4k + 2*lhalf + {0,1}
    const float* a0p = Rp + (size_t)(d0 + l16) * ND + 2 * lhalf;
    const float* a1p = a0p + 16 * ND;
    // B fragment (4x16 f32): row e = 4k + 2*lhalf + v, col = 16c + l16
    const float* fbp = Fp + (size_t)(2 * lhalf) * NB + l16;

    v8f c00 = vzero, c01 = vzero, c10 = vzero, c11 = vzero;

    #pragma unroll 4
    for (int k = 0; k < 64; ++k) {
      v2f a0 = *(const v2f*)(a0p + 4 * k);      // global_load_b64, 2 K-values
      v2f a1 = *(const v2f*)(a1p + 4 * k);
      const float* fk = fbp + (size_t)(4 * k) * NB;
      v2f b0, b1;
      b0.x = fk[0];        b0.y = fk[NB];       // col tile 0, v=0/1
      b1.x = fk[16];       b1.y = fk[NB + 16];  // col tile 1, v=0/1
      // stream-ahead hint on the R panel (global_prefetch_b8)
      __builtin_prefetch(a0p + 4 * k + 64, 0, 0);

      c00 = __builtin_amdgcn_wmma_f32_16x16x4_f32(false, a0, false, b0, (short)0, c00, false, false);
      c01 = __builtin_amdgcn_wmma_f32_16x16x4_f32(false, a0, false, b1, (short)0, c01, false, false);
      c10 = __builtin_amdgcn_wmma_f32_16x16x4_f32(false, a1, false, b0, (short)0, c10, false, false);
      c11 = __builtin_amdgcn_wmma_f32_16x16x4_f32(false, a1, false, b1, (short)0, c11, false, false);
    }
    acc[t][0][0] = c00; acc[t][0][1] = c01;
    acc[t][1][0] = c10; acc[t][1][1] = c11;
  }

  // loss contribution: || M[i,j] - M[j,i] ||^2 for this wave's (d,b) range
  float lsum = 0.f;
  #pragma unroll
  for (int r = 0; r < 2; ++r)
    #pragma unroll
    for (int c = 0; c < 2; ++c)
      #pragma unroll
      for (int v = 0; v < 8; ++v) {
        float dlt = acc[0][r][c][v] - acc[1][r][c][v];
        lsum += dlt * dlt;
      }

  // agg[j] += M[i,j]; agg[i] += M[j,i]
  // C/D element coords: d = d0 + 16r + v + 8*lhalf, b = 16c + l16
  #pragma unroll
  for (int t = 0; t < 2; ++t) {
    const int dst = t ? i : j;
    float* __restrict__ abase = agg + (size_t)dst * (ND * NB);
    #pragma unroll
    for (int r = 0; r < 2; ++r)
      #pragma unroll
      for (int c = 0; c < 2; ++c)
        #pragma unroll
        for (int v = 0; v < 8; ++v) {
          const int d = d0 + 16 * r + v + 8 * lhalf;
          const int b = 16 * c + l16;
          atomicAdd(abase + d * NB + b, acc[t][r][c][v]);
        }
  }

  // block-reduce loss via LDS float atomic, one global atomic per workgroup
  __shared__ float lred;
  if (tid == 0) lred = 0.f;
  __syncthreads();
  atomicAdd(&lred, lsum);
  __syncthreads();
  if (tid == 0) atomicAdd(loss, LAMBDA_W * lred);
}

// ---------------------------------------------------------------------------
// updated = (F + agg) / NP ; append loss scalar.
// ---------------------------------------------------------------------------
__global__ __launch_bounds__(256) void sheaf_finalize_kernel(
    const float* __restrict__ F, const float* __restrict__ agg,
    const float* __restrict__ loss, float* __restrict__ out, int n) {
  int idx = blockIdx.x * blockDim.x + threadIdx.x;
  if (idx < n) out[idx] = (F[idx] + agg[idx]) * (1.0f / (float)NP);
  if (idx == 0) out[n] = loss[0];
}

extern "C" void kernel_launch(void* const* d_in, const int* in_sizes, int n_in,
                              void* d_out, int out_size, void* d_ws, size_t ws_size,
                              hipStream_t stream) {
  const float* F = (const float*)d_in[0];   // [NP, ND, NB]
  const float* R = (const float*)d_in[1];   // [NP, NP, ND, ND]
  float* out = (float*)d_out;               // [NP*ND*NB + 1]
  float* agg = (float*)d_ws;                // [NP*ND*NB]
  const int n = NP * ND * NB;               // 524288
  float* loss = agg + n;

  ws_zero_kernel<<<(n + 1 + 255) / 256, 256, 0, stream>>>(agg, n + 1);
  sheaf_pair_kernel<<<NP * NP, 256, 0, stream>>>(F, R, agg, loss);
  sheaf_finalize_kernel<<<(n + 255) / 256, 256, 0, stream>>>(F, agg, loss, out, n);
}